// GraphEncoder_16363825398375
// MI455X (gfx1250) — compile-verified
//
#include <hip/hip_runtime.h>
#include <hip/hip_bf16.h>
#include <math.h>

#define N_NODES 50000
#define N_EDGES 800000
#define F 64
#define EP (N_EDGES + N_NODES)   // edges + self loops
#define SLOPE_ATT 0.2f
#define SLOPE_ACT 0.01f

typedef __attribute__((ext_vector_type(2))) float v2f;
typedef __attribute__((ext_vector_type(8))) float v8f;

// ---------------------------------------------------------------------------
// h[n,64] = x[n,64] @ W[64,64] using V_WMMA_F32_16X16X4_F32 (full fp32 WMMA).
// One wave computes a 16x64 output strip: 4 N-tiles x 16 K-steps = 64 WMMAs.
// W is staged in LDS (16 KB). Row clamp (not predication) keeps EXEC all-ones
// as WMMA requires; stores are bounds-checked.
// ---------------------------------------------------------------------------
__global__ __launch_bounds__(256) void gat_gemm_wmma(const float* __restrict__ x,
                                                     const float* __restrict__ W,
                                                     float* __restrict__ h, int n) {
  __shared__ float ldsW[F * F];
  for (int i = threadIdx.x; i < (F * F) / 4; i += 256)
    ((float4*)ldsW)[i] = ((const float4*)W)[i];
  __syncthreads();

  const int lane  = threadIdx.x & 31;
  const int wave  = threadIdx.x >> 5;
  const int mBase = (blockIdx.x * 8 + wave) * 16;
  const int r  = lane & 15;   // M (for A) / N (for B,C,D) within tile
  const int hi = lane >> 4;   // lane half selects K pair (A/B) or M+8 (C/D)

  int rowA = mBase + r;
  if (rowA > n - 1) rowA = n - 1;   // clamp so EXEC stays full for WMMA

  v8f acc[4];
  acc[0] = v8f{}; acc[1] = v8f{}; acc[2] = v8f{}; acc[3] = v8f{};

  for (int kt = 0; kt < 16; ++kt) {
    const int kb = kt * 4 + hi * 2;          // A VGPR0/1 -> K=kb, kb+1
    v2f a;
    a.x = x[rowA * F + kb];
    a.y = x[rowA * F + kb + 1];
#pragma unroll
    for (int nt = 0; nt < 4; ++nt) {
      v2f b;                                  // B[k][n] = W[k][n]
      b.x = ldsW[kb * F + nt * 16 + r];
      b.y = ldsW[(kb + 1) * F + nt * 16 + r];
      acc[nt] = __builtin_amdgcn_wmma_f32_16x16x4_f32(
          false, a, false, b, (short)0, acc[nt], false, false);
    }
  }

#pragma unroll
  for (int v = 0; v < 8; ++v) {
    const int row = mBase + v + hi * 8;       // C/D: VGPR v -> M=v (lo half), v+8 (hi half)
    if (row < n) {
#pragma unroll
      for (int nt = 0; nt < 4; ++nt)
        h[row * F + nt * 16 + r] = acc[nt][v];
    }
  }
}

// ---------------------------------------------------------------------------
// alpha_s[i] = h[i].a_src ; alpha_d[i] = h[i].a_dst
// ---------------------------------------------------------------------------
__global__ void gat_alpha(const float* __restrict__ h,
                          const float* __restrict__ a_src,
                          const float* __restrict__ a_dst,
                          float* __restrict__ as, float* __restrict__ ad, int n) {
  int i = blockIdx.x * blockDim.x + threadIdx.x;
  if (i >= n) return;
  const float4* hp = (const float4*)(h + (size_t)i * F);
  float s = 0.f, d = 0.f;
#pragma unroll
  for (int q = 0; q < F / 4; ++q) {
    float4 v  = hp[q];
    float4 vs = ((const float4*)a_src)[q];
    float4 vd = ((const float4*)a_dst)[q];
    s += v.x * vs.x + v.y * vs.y + v.z * vs.z + v.w * vs.w;
    d += v.x * vd.x + v.y * vd.y + v.z * vd.z + v.w * vd.w;
  }
  as[i] = s; ad[i] = d;
}

// ---------------------------------------------------------------------------
// Zero accumulator, init segment-max to -inf, segment-sum to 0.
// ---------------------------------------------------------------------------
__global__ void gat_init(float* __restrict__ accum, float* __restrict__ m,
                         float* __restrict__ z, int n) {
  int j = blockIdx.x * blockDim.x + threadIdx.x;
  if (j < n * F) accum[j] = 0.f;
  if (j < n) { m[j] = -INFINITY; z[j] = 0.f; }
}

__device__ __forceinline__ void atomicMaxF(float* addr, float v) {
  // monotone float-max via sign-split integer atomics (works for mixed signs;
  // init value is -inf = 0xFF800000)
  if (v >= 0.0f) atomicMax((int*)addr, __float_as_int(v));
  else           atomicMin((unsigned int*)addr, __float_as_uint(v));
}

__device__ __forceinline__ void edge_ids(const int* __restrict__ ei, int e,
                                         int& src, int& dst) {
  if (e < N_EDGES) { src = ei[e]; dst = ei[N_EDGES + e]; }
  else             { src = dst = e - N_EDGES; }           // self loops
}

__device__ __forceinline__ float edge_e(const float* as, const float* ad,
                                        int src, int dst) {
  float e = as[src] + ad[dst];
  return e > 0.f ? e : SLOPE_ATT * e;
}

// segment-max of attention logits over destination nodes
__global__ void gat_edge_max(const int* __restrict__ ei,
                             const float* __restrict__ as,
                             const float* __restrict__ ad,
                             float* __restrict__ m) {
  int e = blockIdx.x * blockDim.x + threadIdx.x;
  if (e >= EP) return;
  int src, dst; edge_ids(ei, e, src, dst);
  atomicMaxF(&m[dst], edge_e(as, ad, src, dst));
}

// ex = exp(e - m[dst]); segment-sum into z
__global__ void gat_edge_exp(const int* __restrict__ ei,
                             const float* __restrict__ as,
                             const float* __restrict__ ad,
                             const float* __restrict__ m,
                             float* __restrict__ z, float* __restrict__ exb) {
  int e = blockIdx.x * blockDim.x + threadIdx.x;
  if (e >= EP) return;
  int src, dst; edge_ids(ei, e, src, dst);
  float ex = __expf(edge_e(as, ad, src, dst) - m[dst]);
  exb[e] = ex;
  atomicAdd(&z[dst], ex);
}

// accum[dst] += h[src] * (ex / (z[dst] + 1e-16)); one thread per (edge, feat)
__global__ void gat_edge_agg(const int* __restrict__ ei,
                             const float* __restrict__ h,
                             const float* __restrict__ exb,
                             const float* __restrict__ z,
                             float* __restrict__ accum) {
  long long t = (long long)blockIdx.x * blockDim.x + threadIdx.x;
  if (t >= (long long)EP * F) return;
  int e = (int)(t >> 6);
  int f = (int)(t & 63);
  int src, dst; edge_ids(ei, e, src, dst);
  float alpha = exb[e] / (z[dst] + 1e-16f);
  atomicAdd(&accum[(size_t)dst * F + f], h[(size_t)src * F + f] * alpha);
}

// out = leaky_relu(accum + b, 0.01) [+ residual]
__global__ void gat_finalize(const float* __restrict__ accum,
                             const float* __restrict__ b,
                             const float* __restrict__ xres,  // nullptr => none
                             float* __restrict__ out, int n) {
  int j = blockIdx.x * blockDim.x + threadIdx.x;
  if (j >= n * F) return;
  int f = j & 63;
  float v = accum[j] + b[f];
  v = v > 0.f ? v : SLOPE_ACT * v;
  if (xres) v += xres[j];
  out[j] = v;
}

// ---------------------------------------------------------------------------
extern "C" void kernel_launch(void* const* d_in, const int* in_sizes, int n_in,
                              void* d_out, int out_size, void* d_ws, size_t ws_size,
                              hipStream_t stream) {
  const float* x      = (const float*)d_in[0];
  const int*   ei     = (const int*)d_in[1];
  const float* W0     = (const float*)d_in[2];
  const float* a_src0 = (const float*)d_in[3];
  const float* a_dst0 = (const float*)d_in[4];
  const float* b0     = (const float*)d_in[5];
  const float* W1     = (const float*)d_in[6];
  const float* a_src1 = (const float*)d_in[7];
  const float* a_dst1 = (const float*)d_in[8];
  const float* b1     = (const float*)d_in[9];
  float* out = (float*)d_out;

  // workspace layout (floats): ~42.6 MB total
  float* h     = (float*)d_ws;          // N*F
  float* xmid  = h     + N_NODES * F;   // N*F
  float* accum = xmid  + N_NODES * F;   // N*F
  float* as    = accum + N_NODES * F;   // N
  float* ad    = as    + N_NODES;       // N
  float* m     = ad    + N_NODES;       // N
  float* z     = m     + N_NODES;       // N
  float* exb   = z     + N_NODES;       // EP

  const int B = 256;
  const int gGemm  = (N_NODES + 127) / 128;           // 8 waves x 16 rows / block
  const int gNode  = (N_NODES + B - 1) / B;
  const int gNF    = (N_NODES * F + B - 1) / B;
  const int gEdge  = (EP + B - 1) / B;
  const long long aggT = (long long)EP * F;
  const int gAgg   = (int)((aggT + B - 1) / B);

  // ---- layer 0 ----
  gat_gemm_wmma<<<gGemm, B, 0, stream>>>(x, W0, h, N_NODES);
  gat_alpha    <<<gNode, B, 0, stream>>>(h, a_src0, a_dst0, as, ad, N_NODES);
  gat_init     <<<gNF,   B, 0, stream>>>(accum, m, z, N_NODES);
  gat_edge_max <<<gEdge, B, 0, stream>>>(ei, as, ad, m);
  gat_edge_exp <<<gEdge, B, 0, stream>>>(ei, as, ad, m, z, exb);
  gat_edge_agg <<<gAgg,  B, 0, stream>>>(ei, h, exb, z, accum);
  gat_finalize <<<gNF,   B, 0, stream>>>(accum, b0, nullptr, xmid, N_NODES);

  // ---- layer 1 (+ residual with original x) ----
  gat_gemm_wmma<<<gGemm, B, 0, stream>>>(xmid, W1, h, N_NODES);
  gat_alpha    <<<gNode, B, 0, stream>>>(h, a_src1, a_dst1, as, ad, N_NODES);
  gat_init     <<<gNF,   B, 0, stream>>>(accum, m, z, N_NODES);
  gat_edge_max <<<gEdge, B, 0, stream>>>(ei, as, ad, m);
  gat_edge_exp <<<gEdge, B, 0, stream>>>(ei, as, ad, m, z, exb);
  gat_edge_agg <<<gAgg,  B, 0, stream>>>(ei, h, exb, z, accum);
  gat_finalize <<<gNF,   B, 0, stream>>>(accum, b1, x, out, N_NODES);
}